// UM_loss_1142461301395
// MI455X (gfx1250) — compile-verified
//
#include <hip/hip_runtime.h>
#include <hip/hip_bf16.h>
#include <cstddef>

typedef __attribute__((ext_vector_type(2))) float v2f;
typedef __attribute__((ext_vector_type(8))) float v8f;

#define B_ 64
#define T_ 2048
#define C_ 20
#define K_ 200
#define F_ 2048

// ws layout (floats): [0..63] normsq_act, [64..127] normsq_bkg, [128..1407] bi[b][c]
#define WS_BI 128
#define WS_TOTAL 1408

__device__ __forceinline__ float softplus_pos(float z) {
  // log(1 + exp(z)), numerically stable
  float az = fabsf(z);
  return fmaxf(z, 0.0f) + log1pf(expf(-az));
}

__device__ __forceinline__ float clamped_log(float p) {
  return fmaxf(logf(p), -100.0f); // torch BCELoss log clamp
}

__global__ __launch_bounds__(256) void zero_ws_kernel(float* __restrict__ ws) {
  int i = blockIdx.x * 256 + threadIdx.x;
  if (i < WS_TOTAL) ws[i] = 0.0f;
}

// For each (b, 16-wide f block): sum feat[b, :, f] over K via WMMA f32 16x16x4
// with B = ones (D[m,n] = sum_k A[m,k] independent of B's lane layout),
// then accumulate sum_f (mean^2) into normsq[b].
__global__ __launch_bounds__(256) void feat_norm_kernel(const float* __restrict__ feat,
                                                        float* __restrict__ normsq) {
  const int gwave = (blockIdx.x * 256 + threadIdx.x) >> 5; // 8 waves / block
  const int lane  = threadIdx.x & 31;
  const int b     = gwave >> 7;      // F_/16 = 128 f-blocks per batch row
  const int fblk  = gwave & 127;
  const int m     = lane & 15;       // A-matrix row (= f within block)
  const int half  = lane >> 4;       // lanes 16..31 carry K=2,3 of each row

  // lane reads feat[b, k, fblk*16 + m] for k = 4*i + 2*half + {0,1}
  const float* p = feat + ((size_t)b * K_ + (size_t)(2 * half)) * F_ + (size_t)fblk * 16 + m;

  v8f acc = {0.f, 0.f, 0.f, 0.f, 0.f, 0.f, 0.f, 0.f};
  v2f onesv;
  onesv.x = 1.0f;
  onesv.y = 1.0f;

  for (int i = 0; i < K_ / 4; ++i) {
    v2f a;
    a.x = p[0];    // k = 4i + 2*half
    a.y = p[F_];   // k = 4i + 2*half + 1
    // D = A x ones + C  ->  per-row running sum over K
    acc = __builtin_amdgcn_wmma_f32_16x16x4_f32(false, a, false, onesv,
                                                (short)0, acc, false, false);
    p += 4 * (size_t)F_;
  }

  // D layout: lanes 0-15 hold rows 0..7 (VGPR j <-> M=j), lanes 16-31 rows 8..15.
  // Columns are duplicates (B==1), so only one lane per half contributes.
  if (m == 0) {
    const float invK = 1.0f / (float)K_;
    float s = 0.0f;
#pragma unroll
    for (int j = 0; j < 8; ++j) {
      float mv = acc[j] * invK;
      s += mv * mv;
    }
    atomicAdd(&normsq[b], s);
  }
}

// Per (b,c): pos_mean + neg_mean of the sigmoid BCE over T. Block = (32,8), one block per b.
__global__ __launch_bounds__(256) void bi_kernel(const float* __restrict__ gt,
                                                 const float* __restrict__ cas,
                                                 float* __restrict__ ws_bi) {
  const int b  = blockIdx.x;
  const int c  = threadIdx.x;  // 0..31, active if < C_
  const int ty = threadIdx.y;  // 0..7

  float alp = 0.0f, aln = 0.0f, cnt = 0.0f;
  if (c < C_) {
    const float* gp = gt  + (size_t)b * T_ * C_ + c;
    const float* cp = cas + (size_t)b * T_ * C_ + c;
    for (int t = ty; t < T_; t += 8) {
      float g = gp[(size_t)t * C_];
      float x = cp[(size_t)t * C_];
      float lp = softplus_pos(-x);  // -log sigmoid(x)
      float ln = softplus_pos(x);   // -log(1 - sigmoid(x))
      float pos = (g > 0.5f) ? 1.0f : 0.0f;
      alp += lp * pos;
      aln += ln * (1.0f - pos);
      cnt += pos;
    }
  }

  __shared__ float s_lp[8][32];
  __shared__ float s_ln[8][32];
  __shared__ float s_ct[8][32];
  s_lp[ty][c] = alp;
  s_ln[ty][c] = aln;
  s_ct[ty][c] = cnt;
  __syncthreads();

  if (ty == 0 && c < C_) {
    float SL = 0.0f, SN = 0.0f, SC = 0.0f;
#pragma unroll
    for (int j = 0; j < 8; ++j) {
      SL += s_lp[j][c];
      SN += s_ln[j][c];
      SC += s_ct[j][c];
    }
    float pos_mean = SL / fmaxf(SC, 1.0f);
    float neg_mean = SN / fmaxf((float)T_ - SC, 1.0f);
    ws_bi[b * C_ + c] = pos_mean + 1.0f * neg_mean; // NEG_LMBD = 1
  }
}

// Single block: BCE terms, act-masked bi reduce, UM term, write 6 outputs.
__global__ __launch_bounds__(64) void final_kernel(const float* __restrict__ score_act,
                                                   const float* __restrict__ score_bkg,
                                                   const float* __restrict__ label,
                                                   const float* __restrict__ ws,
                                                   float* __restrict__ out) {
  __shared__ float s_ba[64], s_bb[64], s_as[64], s_ac[64], s_um[64];
  const int b = threadIdx.x;

  float rs = 0.0f;
  for (int c = 0; c < C_; ++c) rs += label[b * C_ + c];
  const float inv_rs = 1.0f / rs; // each row has >= 1 positive

  float bce_a = 0.0f, bce_b = 0.0f, act_sum = 0.0f, act_cnt = 0.0f;
  const float yb = 1.0f / (float)C_;
  for (int c = 0; c < C_; ++c) {
    float lbl = label[b * C_ + c];
    float yn  = lbl * inv_rs;
    float p   = score_act[b * C_ + c];
    bce_a += -(yn * clamped_log(p) + (1.0f - yn) * clamped_log(1.0f - p));
    float q = score_bkg[b * C_ + c];
    bce_b += -(yb * clamped_log(q) + (1.0f - yb) * clamped_log(1.0f - q));
    if (lbl > 0.0f) {
      act_sum += ws[WS_BI + b * C_ + c];
      act_cnt += 1.0f;
    }
  }

  float na = sqrtf(ws[b]);        // ||mean feat_act||
  float nb = sqrtf(ws[64 + b]);   // ||mean feat_bkg||
  float tt = fmaxf(100.0f - na, 0.0f) + nb;

  s_ba[b] = bce_a;
  s_bb[b] = bce_b;
  s_as[b] = act_sum;
  s_ac[b] = act_cnt;
  s_um[b] = tt * tt;
  __syncthreads();

  if (b == 0) {
    float BA = 0.f, BB = 0.f, AS = 0.f, AC = 0.f, UM = 0.f;
    for (int i = 0; i < B_; ++i) {
      BA += s_ba[i]; BB += s_bb[i]; AS += s_as[i]; AC += s_ac[i]; UM += s_um[i];
    }
    const float inv_bc = 1.0f / (float)(B_ * C_);
    float loss_cls = BA * inv_bc;
    float loss_be  = BB * inv_bc;
    float loss_um  = UM / (float)B_;
    float loss_sup_act = 1.0f * (AS / fmaxf(AC, 1.0f)); // LMBD = 1
    float loss_sup_bkg = 0.0f;                           // BKG_LMBD = 0
    float loss_total = loss_cls + 0.0005f * loss_um + 0.2f * loss_be
                       + loss_sup_act + loss_sup_bkg;
    out[0] = loss_total;
    out[1] = loss_cls;
    out[2] = loss_be;
    out[3] = loss_um;
    out[4] = loss_sup_act;
    out[5] = loss_sup_bkg;
  }
}

extern "C" void kernel_launch(void* const* d_in, const int* in_sizes, int n_in,
                              void* d_out, int out_size, void* d_ws, size_t ws_size,
                              hipStream_t stream) {
  (void)in_sizes; (void)n_in; (void)out_size; (void)ws_size;
  const float* score_act = (const float*)d_in[0];
  const float* score_bkg = (const float*)d_in[1];
  const float* feat_act  = (const float*)d_in[2];
  const float* feat_bkg  = (const float*)d_in[3];
  const float* label     = (const float*)d_in[4];
  const float* gt        = (const float*)d_in[5];
  const float* cas       = (const float*)d_in[6];
  float* out = (float*)d_out;
  float* ws  = (float*)d_ws;

  zero_ws_kernel<<<(WS_TOTAL + 255) / 256, 256, 0, stream>>>(ws);
  // 64 b * 128 f-blocks / 8 waves-per-block = 1024 blocks
  feat_norm_kernel<<<1024, 256, 0, stream>>>(feat_act, ws);
  feat_norm_kernel<<<1024, 256, 0, stream>>>(feat_bkg, ws + 64);
  bi_kernel<<<B_, dim3(32, 8), 0, stream>>>(gt, cas, ws + WS_BI);
  final_kernel<<<1, 64, 0, stream>>>(score_act, score_bkg, label, ws, out);
}